// ColorFeatureExtraction_73100343378215
// MI455X (gfx1250) — compile-verified
//
#include <hip/hip_runtime.h>
#include <hip/hip_bf16.h>
#include <math.h>

// Problem constants (from reference setup_inputs)
#define NB   4      // batches
#define NPT  8192   // points per batch

typedef __attribute__((ext_vector_type(2))) float v2f;
typedef __attribute__((ext_vector_type(8))) float v8f;

// Workspace float offsets
#define WS_SUMS 0      // NB*32 = 128 : per-batch channel sums (atomic)
#define WS_CW   128    // NB*32 = 128 : per-batch context gate
#define WS_FW1  256    // 16*4  = 64  : BN-folded W1, K padded 3->4
#define WS_FB1  320    // 16
#define WS_FW2  336    // 32*16 = 512
#define WS_FB2  848    // 32
#define WS_FWA1 880    // 32*32 = 1024
#define WS_FBA1 1904   // 32
#define WS_WA2  1936   // 32*32 = 1024 (no BN; copied for uniform access)
#define WS_BA2  2960   // 32

__device__ __forceinline__ v8f wmma4(v2f a, v2f b, v8f c) {
  // D = A(16x4,f32) x B(4x16,f32) + C(16x16,f32)
  return __builtin_amdgcn_wmma_f32_16x16x4_f32(false, a, false, b,
                                               (short)0, c, false, false);
}

__device__ __forceinline__ float sigmoidf_(float x) {
  return 1.0f / (1.0f + __expf(-x));
}

// ---------------------------------------------------------------------------
// prep: fold BN into 1x1-conv weights, copy wa2/ba2, zero ctx accumulators
// ---------------------------------------------------------------------------
__global__ void prep_kernel(const float* __restrict__ w1, const float* __restrict__ b1,
                            const float* __restrict__ g1, const float* __restrict__ be1,
                            const float* __restrict__ m1, const float* __restrict__ v1,
                            const float* __restrict__ w2, const float* __restrict__ b2,
                            const float* __restrict__ g2, const float* __restrict__ be2,
                            const float* __restrict__ m2, const float* __restrict__ v2,
                            const float* __restrict__ wa1, const float* __restrict__ ba1,
                            const float* __restrict__ ga, const float* __restrict__ bea,
                            const float* __restrict__ ma, const float* __restrict__ va,
                            const float* __restrict__ wa2, const float* __restrict__ ba2,
                            float* __restrict__ ws) {
  const int t = threadIdx.x;
  for (int i = t; i < NB * 32; i += blockDim.x) ws[WS_SUMS + i] = 0.0f;

  for (int i = t; i < 16 * 4; i += blockDim.x) {          // W1 folded, K pad
    int o = i >> 2, c = i & 3;
    float s = g1[o] * rsqrtf(v1[o] + 1e-5f);
    ws[WS_FW1 + i] = (c < 3) ? w1[o * 3 + c] * s : 0.0f;
  }
  for (int o = t; o < 16; o += blockDim.x) {
    float s = g1[o] * rsqrtf(v1[o] + 1e-5f);
    ws[WS_FB1 + o] = (b1[o] - m1[o]) * s + be1[o];
  }
  for (int i = t; i < 32 * 16; i += blockDim.x) {         // W2 folded
    int o = i >> 4;
    float s = g2[o] * rsqrtf(v2[o] + 1e-5f);
    ws[WS_FW2 + i] = w2[i] * s;
  }
  for (int o = t; o < 32; o += blockDim.x) {
    float s = g2[o] * rsqrtf(v2[o] + 1e-5f);
    ws[WS_FB2 + o] = (b2[o] - m2[o]) * s + be2[o];
  }
  for (int i = t; i < 32 * 32; i += blockDim.x) {         // Wa1 folded
    int o = i >> 5;
    float s = ga[o] * rsqrtf(va[o] + 1e-5f);
    ws[WS_FWA1 + i] = wa1[i] * s;
  }
  for (int o = t; o < 32; o += blockDim.x) {
    float s = ga[o] * rsqrtf(va[o] + 1e-5f);
    ws[WS_FBA1 + o] = (ba1[o] - ma[o]) * s + bea[o];
  }
  for (int i = t; i < 32 * 32; i += blockDim.x) ws[WS_WA2 + i] = wa2[i];
  for (int o = t; o < 32; o += blockDim.x) ws[WS_BA2 + o] = ba2[o];
}

// ---------------------------------------------------------------------------
// Per-wave color MLP 3 -> 16 -> 32 on a 16-point tile, fp32 WMMA.
// bufF: per-wave LDS region, >= 512 floats. Result in flo/fhi:
//   register j, lanes 0-15:  channel j+8*half(+16 for fhi) ... col = point.
// ---------------------------------------------------------------------------
__device__ __forceinline__ void compute_f(const float* __restrict__ colors,
                                          const float* __restrict__ ws,
                                          float* bufF,
                                          int b, int n0, int col, int half,
                                          v8f& flo, v8f& fhi) {
  const float* colB = colors + (size_t)b * 3 * NPT;

  // ---- layer1: 3->16 (K padded to 4), one WMMA ----
  v2f bf, a1;
  bf.x = colB[(2 * half) * NPT + n0 + col];                 // K row 0 / 2
  bf.y = half ? 0.0f : colB[NPT + n0 + col];                // K row 1 / 3(pad)
  a1.x = ws[WS_FW1 + col * 4 + 2 * half];
  a1.y = ws[WS_FW1 + col * 4 + 2 * half + 1];
  v8f f1 = {};
  f1 = wmma4(a1, bf, f1);
#pragma unroll
  for (int j = 0; j < 8; j++) {
    int o = j + 8 * half;
    f1[j] = fmaxf(f1[j] + ws[WS_FB1 + o], 0.0f);
    bufF[o * 16 + col] = f1[j];                             // stage 16x16 tile
  }

  // ---- layer2: 16->32, K=16 in 4 chunks, two 16-row output tiles ----
  v8f lo = {}, hi = {};
#pragma unroll
  for (int q = 0; q < 4; q++) {
    int kc = 4 * q + 2 * half;
    v2f bb; bb.x = bufF[kc * 16 + col]; bb.y = bufF[(kc + 1) * 16 + col];
    v2f al; al.x = ws[WS_FW2 + col * 16 + kc];
            al.y = ws[WS_FW2 + col * 16 + kc + 1];
    v2f ah; ah.x = ws[WS_FW2 + (16 + col) * 16 + kc];
            ah.y = ws[WS_FW2 + (16 + col) * 16 + kc + 1];
    lo = wmma4(al, bb, lo);
    hi = wmma4(ah, bb, hi);
  }
#pragma unroll
  for (int j = 0; j < 8; j++) {
    lo[j] = fmaxf(lo[j] + ws[WS_FB2 + j + 8 * half], 0.0f);
    hi[j] = fmaxf(hi[j] + ws[WS_FB2 + 16 + j + 8 * half], 0.0f);
  }
  flo = lo; fhi = hi;
}

// ---------------------------------------------------------------------------
// Pass 1: channel sums of f for the global-context branch
// ---------------------------------------------------------------------------
__global__ __launch_bounds__(256) void sum_kernel(const float* __restrict__ colors,
                                                  float* __restrict__ ws) {
  __shared__ float lds[8 * 512];                            // 8 waves x [32][16]
  const int t = threadIdx.x;
  const int wave = t >> 5, lane = t & 31;
  const int col = lane & 15, half = lane >> 4;
  const int tilesPerBatch = NPT / 128;
  const int b = blockIdx.x / tilesPerBatch;
  const int n0 = (blockIdx.x % tilesPerBatch) * 128 + wave * 16;
  float* bufF = lds + wave * 512;

  v8f flo, fhi;
  compute_f(colors, ws, bufF, b, n0, col, half, flo, fhi);

#pragma unroll
  for (int j = 0; j < 8; j++) {                             // stage full 32x16 f
    bufF[(j + 8 * half) * 16 + col] = flo[j];
    bufF[(16 + j + 8 * half) * 16 + col] = fhi[j];
  }
  __syncthreads();

  if (t < 32) {                                             // channel t over 128 pts
    float s = 0.0f;
    for (int w = 0; w < 8; w++)
#pragma unroll
      for (int p = 0; p < 16; p++) s += lds[w * 512 + t * 16 + p];
    atomicAdd(&ws[WS_SUMS + b * 32 + t], s);
  }
}

// ---------------------------------------------------------------------------
// Pass 2: squeeze-excite global gate cw = sigmoid(Wc2 relu(Wc1 mean + bc1) + bc2)
// ---------------------------------------------------------------------------
__global__ void ctx_kernel(const float* __restrict__ wc1, const float* __restrict__ bc1,
                           const float* __restrict__ wc2, const float* __restrict__ bc2,
                           float* __restrict__ ws) {
  __shared__ float mch[32];
  __shared__ float h[16];
  const int b = blockIdx.x, t = threadIdx.x;
  if (t < 32) mch[t] = ws[WS_SUMS + b * 32 + t] * (1.0f / (float)NPT);
  __syncthreads();
  if (t < 16) {
    float acc = bc1[t];
    for (int c = 0; c < 32; c++) acc += wc1[t * 32 + c] * mch[c];
    h[t] = fmaxf(acc, 0.0f);
  }
  __syncthreads();
  if (t < 32) {
    float acc = bc2[t];
    for (int c = 0; c < 16; c++) acc += wc2[t * 16 + c] * h[c];
    ws[WS_CW + b * 32 + t] = sigmoidf_(acc);
  }
}

// ---------------------------------------------------------------------------
// Pass 3: recompute f, channel attention (two 32x32 WMMA layers), final gate,
// coalesced store of out = f * sigmoid(attn) * cw
// ---------------------------------------------------------------------------
__global__ __launch_bounds__(256) void fuse_kernel(const float* __restrict__ colors,
                                                   const float* __restrict__ ws,
                                                   float* __restrict__ out) {
  __shared__ float lds[8 * 1024];                           // 8 waves x (F 512 + T 512)
  const int t = threadIdx.x;
  const int wave = t >> 5, lane = t & 31;
  const int col = lane & 15, half = lane >> 4;
  const int tilesPerBatch = NPT / 128;
  const int b = blockIdx.x / tilesPerBatch;
  const int n0 = (blockIdx.x % tilesPerBatch) * 128 + wave * 16;
  float* bufF = lds + wave * 1024;
  float* bufT = bufF + 512;

  v8f flo, fhi;
  compute_f(colors, ws, bufF, b, n0, col, half, flo, fhi);

#pragma unroll
  for (int j = 0; j < 8; j++) {                             // stage f as [32][16]
    bufF[(j + 8 * half) * 16 + col] = flo[j];
    bufF[(16 + j + 8 * half) * 16 + col] = fhi[j];
  }

  // ---- attention layer 1: 32->32 (BN folded), K=32 in 8 chunks ----
  v8f tlo = {}, thi = {};
#pragma unroll
  for (int q = 0; q < 8; q++) {
    int kc = 4 * q + 2 * half;
    v2f bb; bb.x = bufF[kc * 16 + col]; bb.y = bufF[(kc + 1) * 16 + col];
    v2f al; al.x = ws[WS_FWA1 + col * 32 + kc];
            al.y = ws[WS_FWA1 + col * 32 + kc + 1];
    v2f ah; ah.x = ws[WS_FWA1 + (16 + col) * 32 + kc];
            ah.y = ws[WS_FWA1 + (16 + col) * 32 + kc + 1];
    tlo = wmma4(al, bb, tlo);
    thi = wmma4(ah, bb, thi);
  }
#pragma unroll
  for (int j = 0; j < 8; j++) {                             // bias+relu, stage [32][16]
    float a = fmaxf(tlo[j] + ws[WS_FBA1 + j + 8 * half], 0.0f);
    float c = fmaxf(thi[j] + ws[WS_FBA1 + 16 + j + 8 * half], 0.0f);
    bufT[(j + 8 * half) * 16 + col] = a;
    bufT[(16 + j + 8 * half) * 16 + col] = c;
  }

  // ---- attention layer 2: 32->32, pre-sigmoid logits ----
  v8f ulo = {}, uhi = {};
#pragma unroll
  for (int q = 0; q < 8; q++) {
    int kc = 4 * q + 2 * half;
    v2f bb; bb.x = bufT[kc * 16 + col]; bb.y = bufT[(kc + 1) * 16 + col];
    v2f al; al.x = ws[WS_WA2 + col * 32 + kc];
            al.y = ws[WS_WA2 + col * 32 + kc + 1];
    v2f ah; ah.x = ws[WS_WA2 + (16 + col) * 32 + kc];
            ah.y = ws[WS_WA2 + (16 + col) * 32 + kc + 1];
    ulo = wmma4(al, bb, ulo);
    uhi = wmma4(ah, bb, uhi);
  }

  // ---- gate and store: out[b][o][n] = f * sigmoid(u + ba2) * cw[b][o] ----
  const float* cw = ws + WS_CW + b * 32;
  float* ob = out + (size_t)b * 32 * NPT;
#pragma unroll
  for (int j = 0; j < 8; j++) {
    int olo = j + 8 * half, ohi = 16 + j + 8 * half;
    float glo = sigmoidf_(ulo[j] + ws[WS_BA2 + olo]);
    float ghi = sigmoidf_(uhi[j] + ws[WS_BA2 + ohi]);
    ob[(size_t)olo * NPT + n0 + col] = flo[j] * glo * cw[olo];
    ob[(size_t)ohi * NPT + n0 + col] = fhi[j] * ghi * cw[ohi];
  }
}

// ---------------------------------------------------------------------------
extern "C" void kernel_launch(void* const* d_in, const int* in_sizes, int n_in,
                              void* d_out, int out_size, void* d_ws, size_t ws_size,
                              hipStream_t stream) {
  const float* colors = (const float*)d_in[0];
  // d_in[1] = xyz : DEAD in the reference (kNN result is deleted) -> skipped.
  const float* w1 = (const float*)d_in[2];  const float* b1 = (const float*)d_in[3];
  const float* g1 = (const float*)d_in[4];  const float* be1 = (const float*)d_in[5];
  const float* m1 = (const float*)d_in[6];  const float* v1 = (const float*)d_in[7];
  const float* w2 = (const float*)d_in[8];  const float* b2 = (const float*)d_in[9];
  const float* g2 = (const float*)d_in[10]; const float* be2 = (const float*)d_in[11];
  const float* m2 = (const float*)d_in[12]; const float* v2 = (const float*)d_in[13];
  const float* wa1 = (const float*)d_in[14]; const float* ba1 = (const float*)d_in[15];
  const float* ga  = (const float*)d_in[16]; const float* bea = (const float*)d_in[17];
  const float* ma  = (const float*)d_in[18]; const float* va  = (const float*)d_in[19];
  const float* wa2 = (const float*)d_in[20]; const float* ba2 = (const float*)d_in[21];
  const float* wc1 = (const float*)d_in[22]; const float* bc1 = (const float*)d_in[23];
  const float* wc2 = (const float*)d_in[24]; const float* bc2 = (const float*)d_in[25];

  float* ws  = (float*)d_ws;
  float* out = (float*)d_out;

  const int blocks = NB * (NPT / 128);   // 256 blocks, 8 waves each

  prep_kernel<<<1, 256, 0, stream>>>(w1, b1, g1, be1, m1, v1,
                                     w2, b2, g2, be2, m2, v2,
                                     wa1, ba1, ga, bea, ma, va,
                                     wa2, ba2, ws);
  sum_kernel<<<blocks, 256, 0, stream>>>(colors, ws);
  ctx_kernel<<<NB, 32, 0, stream>>>(wc1, bc1, wc2, bc2, ws);
  fuse_kernel<<<blocks, 256, 0, stream>>>(colors, ws, out);
}